// GraphAttentionLayer_10479720202952
// MI455X (gfx1250) — compile-verified
//
#include <hip/hip_runtime.h>
#include <hip/hip_bf16.h>

#define N_NODES 8192
#define N_EDGES 262144
#define IN_F    256
#define OUT_F   64
#define ALPHA   0.2f

typedef __attribute__((ext_vector_type(2))) float v2f;
typedef __attribute__((ext_vector_type(8))) float v8f;

__device__ __forceinline__ float leaky(float x) {
    return x >= 0.0f ? x : ALPHA * x;
}

// Deterministic float atomic max via monotone int encoding.
__device__ __forceinline__ void atomicMaxFloat(float* addr, float v) {
    if (v >= 0.0f)
        atomicMax((int*)addr, __float_as_int(v));
    else
        atomicMin((unsigned int*)addr, (unsigned int)__float_as_int(v));
}

// ---------------------------------------------------------------------------
// K0: init rowmax = -inf, rowsum = 0
// ---------------------------------------------------------------------------
__global__ void gat_init_rows(float* __restrict__ rowmax, float* __restrict__ rowsum) {
    int i = blockIdx.x * blockDim.x + threadIdx.x;
    if (i < N_NODES) {
        rowmax[i] = -__builtin_huge_valf();
        rowsum[i] = 0.0f;
    }
}

// ---------------------------------------------------------------------------
// K1: Wh = h @ W via V_WMMA_F32_16X16X4_F32.
// Grid: 256 blocks x 256 threads (8 waves). Wave w in block b computes the
// 16x16 tile (mtile = 2b + w/4, ntile = w%4). W staged through LDS in
// 64-row (16 KB) chunks.
// ---------------------------------------------------------------------------
__global__ __launch_bounds__(256)
void gat_gemm_wmma(const float* __restrict__ h, const float* __restrict__ W,
                   float* __restrict__ Wh) {
    __shared__ float Wlds[64 * OUT_F];   // 16 KB chunk of W

    const int lane = threadIdx.x & 31;
    const int wave = threadIdx.x >> 5;
    const int half = lane >> 4;          // 0: lanes 0-15, 1: lanes 16-31
    const int l    = lane & 15;

    const int mtile = blockIdx.x * 2 + (wave >> 2);
    const int ntile = wave & 3;
    const int m0 = mtile * 16;
    const int n0 = ntile * 16;
    const int row = m0 + l;              // A row for this lane
    const int col = n0 + l;              // B column for this lane

    v8f c = {};

    for (int kc = 0; kc < IN_F; kc += 64) {
        // Cooperative, coalesced stage of W[kc:kc+64, :] into LDS.
        for (int idx = threadIdx.x; idx < 64 * OUT_F; idx += 256)
            Wlds[idx] = W[kc * OUT_F + idx];
        __syncthreads();

        for (int k0 = 0; k0 < 64; k0 += 4) {
            const int kA = kc + k0 + 2 * half;
            // Hint the next chunk of this A row into cache.
            __builtin_prefetch(&h[(size_t)row * IN_F + kA + 64], 0, 1);

            // A fragment: lane holds h[row][kA], h[row][kA+1]  (global_load_b64)
            v2f a = *reinterpret_cast<const v2f*>(&h[(size_t)row * IN_F + kA]);

            // B fragment from LDS: rows kB, kB+1 of W chunk at column `col`
            const int kB = k0 + 2 * half;
            v2f b;
            b.x = Wlds[kB * OUT_F + col];
            b.y = Wlds[(kB + 1) * OUT_F + col];

            c = __builtin_amdgcn_wmma_f32_16x16x4_f32(
                    false, a, false, b, (short)0, c, false, false);
        }
        __syncthreads();
    }

    // C/D layout: VGPR v -> M = m0 + 8*half + v, N = n0 + l
    #pragma unroll
    for (int v = 0; v < 8; ++v)
        Wh[(size_t)(m0 + 8 * half + v) * OUT_F + n0 + l] = c[v];
}

// ---------------------------------------------------------------------------
// K2: per-node scores s1 = Wh@a1, s2 = Wh@a2. One wave32 per node.
// ---------------------------------------------------------------------------
__global__ __launch_bounds__(256)
void gat_scores(const float* __restrict__ Wh, const float* __restrict__ a,
                float* __restrict__ s1, float* __restrict__ s2) {
    const int lane = threadIdx.x & 31;
    const int node = blockIdx.x * 8 + (threadIdx.x >> 5);
    if (node >= N_NODES) return;

    const float* whr = Wh + (size_t)node * OUT_F;
    float p1 = whr[lane] * a[lane] + whr[lane + 32] * a[lane + 32];
    float p2 = whr[lane] * a[OUT_F + lane] + whr[lane + 32] * a[OUT_F + lane + 32];

    #pragma unroll
    for (int off = 16; off > 0; off >>= 1) {
        p1 += __shfl_xor(p1, off, 32);
        p2 += __shfl_xor(p2, off, 32);
    }
    if (lane == 0) {
        s1[node] = p1;
        s2[node] = p2;
    }
}

// ---------------------------------------------------------------------------
// K2b: column means of Wh (fallback for edgeless rows). One block per column.
// ---------------------------------------------------------------------------
__global__ __launch_bounds__(256)
void gat_colmean(const float* __restrict__ Wh, float* __restrict__ meanWh) {
    __shared__ float red[256];
    const int f = blockIdx.x;
    float s = 0.0f;
    for (int i = threadIdx.x; i < N_NODES; i += 256)
        s += Wh[(size_t)i * OUT_F + f];
    red[threadIdx.x] = s;
    __syncthreads();
    for (int off = 128; off > 0; off >>= 1) {
        if (threadIdx.x < off) red[threadIdx.x] += red[threadIdx.x + off];
        __syncthreads();
    }
    if (threadIdx.x == 0) meanWh[f] = red[0] / (float)N_NODES;
}

// ---------------------------------------------------------------------------
// K3: edge pass 1 — dedupe via bitmap, record first-occurrence flag,
// accumulate deterministic row max of leaky(s1[src]+s2[dst]).
// ---------------------------------------------------------------------------
__global__ __launch_bounds__(256)
void gat_edge_max(const int* __restrict__ src, const int* __restrict__ dst,
                  const float* __restrict__ s1, const float* __restrict__ s2,
                  unsigned int* __restrict__ bitmap,
                  unsigned char* __restrict__ flags,
                  float* __restrict__ rowmax) {
    const int e = blockIdx.x * blockDim.x + threadIdx.x;
    if (e >= N_EDGES) return;
    const int s = src[e];
    const int d = dst[e];
    const unsigned long long bitpos = (unsigned long long)s * N_NODES + d;
    const unsigned int word = (unsigned int)(bitpos >> 5);
    const unsigned int bit  = 1u << (bitpos & 31);
    const unsigned int old  = atomicOr(&bitmap[word], bit);
    const bool first = (old & bit) == 0;
    flags[e] = first ? 1 : 0;
    if (first) {
        const float el = leaky(s1[s] + s2[d]);
        atomicMaxFloat(&rowmax[s], el);
    }
}

// ---------------------------------------------------------------------------
// K4: edge pass 2 — softmax numerators + scatter-accumulate attention @ Wh.
// 8 threads per edge, each handling 8 contiguous output features.
// ---------------------------------------------------------------------------
__global__ __launch_bounds__(256)
void gat_edge_acc(const int* __restrict__ src, const int* __restrict__ dst,
                  const float* __restrict__ s1, const float* __restrict__ s2,
                  const float* __restrict__ rowmax,
                  const unsigned char* __restrict__ flags,
                  const float* __restrict__ Wh,
                  float* __restrict__ rowsum, float* __restrict__ acc) {
    const int gid  = blockIdx.x * blockDim.x + threadIdx.x;
    const int e    = gid >> 3;
    const int part = gid & 7;
    if (e >= N_EDGES || !flags[e]) return;

    const int s = src[e];
    const int d = dst[e];
    const float el = leaky(s1[s] + s2[d]);
    const float w  = __expf(el - rowmax[s]);

    if (part == 0) atomicAdd(&rowsum[s], w);

    const float* whd = Wh + (size_t)d * OUT_F + part * 8;
    float* out = acc + (size_t)s * OUT_F + part * 8;
    #pragma unroll
    for (int j = 0; j < 8; ++j)
        atomicAdd(&out[j], w * whd[j]);
}

// ---------------------------------------------------------------------------
// K5: finalize — normalize, handle empty rows (uniform attention => column
// mean), apply leaky_relu.
// ---------------------------------------------------------------------------
__global__ __launch_bounds__(256)
void gat_finalize(float* __restrict__ out, const float* __restrict__ rowsum,
                  const float* __restrict__ meanWh) {
    const int idx = blockIdx.x * blockDim.x + threadIdx.x;
    if (idx >= N_NODES * OUT_F) return;
    const float sum = rowsum[idx >> 6];
    const float val = (sum > 0.0f) ? out[idx] / sum : meanWh[idx & 63];
    out[idx] = leaky(val);
}

// ---------------------------------------------------------------------------
// Workspace layout (bytes, all 256-aligned)
// ---------------------------------------------------------------------------
#define WH_OFF     ((size_t)0)                        // 8192*64*4  = 2097152
#define S1_OFF     ((size_t)2097152)                  // 32768
#define S2_OFF     ((size_t)2129920)                  // 32768
#define RMAX_OFF   ((size_t)2162688)                  // 32768
#define RSUM_OFF   ((size_t)2195456)                  // 32768
#define MEAN_OFF   ((size_t)2228224)                  // 256
#define FLAGS_OFF  ((size_t)2228480)                  // 262144
#define BITMAP_OFF ((size_t)2490624)                  // 8388608
#define WS_NEEDED  ((size_t)10879232)

extern "C" void kernel_launch(void* const* d_in, const int* in_sizes, int n_in,
                              void* d_out, int out_size, void* d_ws, size_t ws_size,
                              hipStream_t stream) {
    const float* h = (const float*)d_in[0];
    const float* W = (const float*)d_in[1];
    const float* a = (const float*)d_in[2];
    const int* src = (const int*)d_in[3];
    const int* dst = (const int*)d_in[4];
    float* out = (float*)d_out;

    char* ws = (char*)d_ws;
    float*         Wh     = (float*)(ws + WH_OFF);
    float*         s1     = (float*)(ws + S1_OFF);
    float*         s2     = (float*)(ws + S2_OFF);
    float*         rowmax = (float*)(ws + RMAX_OFF);
    float*         rowsum = (float*)(ws + RSUM_OFF);
    float*         meanWh = (float*)(ws + MEAN_OFF);
    unsigned char* flags  = (unsigned char*)(ws + FLAGS_OFF);
    unsigned int*  bitmap = (unsigned int*)(ws + BITMAP_OFF);

    // Zero the dedup bitmap and the output accumulator (graph-capture safe).
    hipMemsetAsync(bitmap, 0, (size_t)N_NODES * N_NODES / 8, stream);
    hipMemsetAsync(out, 0, (size_t)N_NODES * OUT_F * sizeof(float), stream);

    gat_init_rows<<<N_NODES / 256, 256, 0, stream>>>(rowmax, rowsum);

    // Wh = h @ W (WMMA f32 16x16x4). 512 M-tiles x 4 N-tiles; 8 waves/block.
    gat_gemm_wmma<<<256, 256, 0, stream>>>(h, W, Wh);

    gat_scores<<<N_NODES / 8, 256, 0, stream>>>(Wh, a, s1, s2);
    gat_colmean<<<OUT_F, 256, 0, stream>>>(Wh, meanWh);

    gat_edge_max<<<N_EDGES / 256, 256, 0, stream>>>(src, dst, s1, s2,
                                                    bitmap, flags, rowmax);

    gat_edge_acc<<<(N_EDGES * 8) / 256, 256, 0, stream>>>(src, dst, s1, s2,
                                                          rowmax, flags, Wh,
                                                          rowsum, out);

    gat_finalize<<<(N_NODES * OUT_F) / 256, 256, 0, stream>>>(out, rowsum, meanWh);
}